// SemanticClassifier_85023172592344
// MI455X (gfx1250) — compile-verified
//
#include <hip/hip_runtime.h>
#include <hip/hip_bf16.h>

typedef float v2f __attribute__((ext_vector_type(2)));
typedef float v8f __attribute__((ext_vector_type(8)));

#define POOL 5
#define PQ   25
#define CC   512
#define HH   22
#define WW   22
#define HW   484          // 22*22
#define NG   121          // HW / 4 K-groups
#define CK   12800        // CC*PQ
#define NROI 256
#define RSCALE (1.0f/16.0f)

__device__ __forceinline__ float hat_int(float t) {
    // antiderivative of max(0, 1-|u|), H(-1)=0, H(1)=1
    t = fminf(1.0f, fmaxf(-1.0f, t));
    float a = t + 1.0f, b = 1.0f - t;
    return (t < 0.0f) ? 0.5f * a * a : 1.0f - 0.5f * b * b;
}

__device__ __forceinline__ v8f wmma_f32(v2f a, v2f b, v8f c) {
    return __builtin_amdgcn_wmma_f32_16x16x4_f32(false, a, false, b,
                                                 (short)0, c, false, false);
}

// Kernel 1: precise ROI pooling as dense GEMM per ROI:
//   pooled[n][c][pq] = sum_hw feat[n][c][hw] * W2[pq][hw]
// One block per ROI. W2 is built in LDS directly in WMMA B-operand lane order:
//   sB[(bt*NG + g)*32 + lane] = { W2[pq][k], W2[pq][k+1] },
//   pq = bt*16 + (lane&15), k = 4g + 2*(lane>>4)   (bank-conflict-free reads)
__global__ __launch_bounds__(256) void prroi_pool_wmma(
    const float* __restrict__ feat, const float* __restrict__ bb,
    float* __restrict__ pooled)
{
    __shared__ v2f   sB[2 * NG * 32];       // 61.9 KB
    __shared__ float sWx[POOL * WW];
    __shared__ float sWy[POOL * HH];

    const int n   = blockIdx.x;
    const int tid = threadIdx.x;

    // ROI scalars (uniform; bb is tiny and L2-hot)
    const float bx = bb[n*4+0], by = bb[n*4+1], bwid = bb[n*4+2], bhgt = bb[n*4+3];
    const float x1 = bx * RSCALE, y1 = by * RSCALE;
    const float x2 = (bx + bwid) * RSCALE, y2 = (by + bhgt) * RSCALE;
    const float bw = fmaxf((x2 - x1) * (1.0f/POOL), 0.0f);
    const float bh = fmaxf((y2 - y1) * (1.0f/POOL), 0.0f);
    const float area = bw * bh;
    const float inv_area = (area > 0.0f) ? (1.0f / fmaxf(area, 1e-12f)) : 0.0f;

    if (tid < POOL*WW) {
        int q = tid / WW, i = tid % WW;
        sWx[tid] = hat_int(x1 + (q+1)*bw - (float)i) - hat_int(x1 + q*bw - (float)i);
    } else if (tid < POOL*WW + POOL*HH) {
        int t2 = tid - POOL*WW;
        int p = t2 / HH, j = t2 % HH;
        sWy[t2] = hat_int(y1 + (p+1)*bh - (float)j) - hat_int(y1 + p*bh - (float)j);
    }
    __syncthreads();

    for (int idx = tid; idx < 2*NG*32; idx += 256) {
        int bt = idx / (NG*32);
        int r  = idx - bt*NG*32;
        int g  = r >> 5;
        int ln = r & 31;
        int pq = bt*16 + (ln & 15);
        int k  = 4*g + 2*(ln >> 4);
        v2f val = {0.0f, 0.0f};
        if (pq < PQ) {
            int p = pq / POOL, q = pq % POOL;
            float wp0 = sWy[p*HH + (k    / WW)] * sWx[q*WW + (k    % WW)];
            float wp1 = sWy[p*HH + ((k+1)/ WW)] * sWx[q*WW + ((k+1)% WW)];
            val.x = wp0 * inv_area;
            val.y = wp1 * inv_area;
        }
        sB[idx] = val;
    }
    __syncthreads();

    const int wv   = tid >> 5;
    const int lane = tid & 31;
    const int m16  = lane & 15;          // A: M row / B: column
    const int koff = (lane >> 4) * 2;    // lanes 16-31 carry K=2,3

    const float* fn = feat + (long)n * (CC * HW);
    float*       pn = pooled + (long)n * CK;

    const v2f* B0 = sB + lane;           // bins 0..15
    const v2f* B1 = sB + NG*32 + lane;   // bins 16..31 (zero-padded)

    for (int job = 0; job < 4; ++job) {
        const int ctile = wv + job * 8;                // 32 channel tiles / 8 waves
        const float* ap = fn + (long)(ctile*16 + m16) * HW + koff;

        v8f acc0 = {};
        v8f acc1 = {};
        // 4 K-groups per iteration: 4 independent global loads issued together
        #pragma unroll 1
        for (int g = 0; g < NG - 1; g += 4) {
            const float* base = ap + g*4;
            v2f x0 = *(const v2f*)(base + 0);
            v2f x1 = *(const v2f*)(base + 4);
            v2f x2 = *(const v2f*)(base + 8);
            v2f x3 = *(const v2f*)(base + 12);
            const v2f* pb0 = B0 + g*32;
            const v2f* pb1 = B1 + g*32;
            acc0 = wmma_f32(x0, pb0[ 0], acc0);  acc1 = wmma_f32(x0, pb1[ 0], acc1);
            acc0 = wmma_f32(x1, pb0[32], acc0);  acc1 = wmma_f32(x1, pb1[32], acc1);
            acc0 = wmma_f32(x2, pb0[64], acc0);  acc1 = wmma_f32(x2, pb1[64], acc1);
            acc0 = wmma_f32(x3, pb0[96], acc0);  acc1 = wmma_f32(x3, pb1[96], acc1);
        }
        {   // epilogue: group 120 (kk = 480)
            v2f x = *(const v2f*)(ap + 480);
            acc0 = wmma_f32(x, B0[120*32], acc0);
            acc1 = wmma_f32(x, B1[120*32], acc1);
        }

        // D layout: lane -> column ncol = lane&15, rows 8*(lane>>4)+j
        const int ncol  = m16;
        const int mbase = (lane >> 4) * 8;
        float* p0 = pn + ncol;
        #pragma unroll
        for (int j = 0; j < 8; ++j) p0[(ctile*16 + mbase + j) * PQ] = acc0[j];
        if (ncol < 9) {                      // single predicate for second bin tile
            float* p1 = pn + 16 + ncol;
            #pragma unroll
            for (int j = 0; j < 8; ++j) p1[(ctile*16 + mbase + j) * PQ] = acc1[j];
        }
    }
}

// Kernel 2: heads as one GEMM  out[roi, nh] = pooled[roi,:] . wcat[nh,:]
// M = 256 rois (16 tiles), N = 24 heads (padded to 2x16), K = 12800 split over 8 waves.
__global__ __launch_bounds__(256) void head_wmma(
    const float* __restrict__ pooled,
    const float* __restrict__ w_obj, const float* __restrict__ b_obj,
    const float* __restrict__ w_mot, const float* __restrict__ b_mot,
    float* __restrict__ out)
{
    __shared__ float red[8 * 256];
    const int mtile = blockIdx.x;       // 16
    const int ntile = blockIdx.y;       // 2
    const int tid   = threadIdx.x;
    const int wv    = tid >> 5;
    const int lane  = tid & 31;
    const int m16   = lane & 15;
    const int kpair = (lane >> 4) * 2;
    const int KSL   = CK / 8;           // 1600 per wave
    const int GSL   = KSL / 4;          // 400 K-groups per wave

    // B column nh: rows of the (virtually concatenated) weight matrix
    const int nh = ntile*16 + m16;
    const float* wrow;
    if      (nh < 20) wrow = w_obj + (long)nh * CK;
    else if (nh < 24) wrow = w_mot + (long)(nh - 20) * CK;
    else              wrow = w_obj;     // safe dummy; column never stored

    const float* ap = pooled + (long)(mtile*16 + m16) * CK + kpair + wv * KSL;
    const float* bp = wrow + kpair + wv * KSL;

    v8f acc = {};
    // 4 K-groups per iteration: 8 independent global loads issued together
    #pragma unroll 1
    for (int g = 0; g < GSL; g += 4) {
        const float* a4 = ap + g*4;
        const float* b4 = bp + g*4;
        v2f xa0 = *(const v2f*)(a4 + 0);
        v2f xa1 = *(const v2f*)(a4 + 4);
        v2f xa2 = *(const v2f*)(a4 + 8);
        v2f xa3 = *(const v2f*)(a4 + 12);
        v2f xb0 = *(const v2f*)(b4 + 0);
        v2f xb1 = *(const v2f*)(b4 + 4);
        v2f xb2 = *(const v2f*)(b4 + 8);
        v2f xb3 = *(const v2f*)(b4 + 12);
        acc = wmma_f32(xa0, xb0, acc);
        acc = wmma_f32(xa1, xb1, acc);
        acc = wmma_f32(xa2, xb2, acc);
        acc = wmma_f32(xa3, xb3, acc);
    }

    #pragma unroll
    for (int j = 0; j < 8; ++j) red[wv*256 + lane*8 + j] = acc[j];
    __syncthreads();

    float s = 0.0f;
    #pragma unroll
    for (int w = 0; w < 8; ++w) s += red[w*256 + tid];

    const int lsrc = tid >> 3, j = tid & 7;
    const int ncol = lsrc & 15;
    const int mrow = (lsrc >> 4) * 8 + j;
    const int nh2  = ntile*16 + ncol;
    const int roi  = mtile*16 + mrow;
    if (nh2 < 24) {
        float bias = (nh2 < 20) ? b_obj[nh2] : b_mot[nh2 - 20];
        float val = s + bias;
        if (nh2 < 20) out[roi*20 + nh2] = val;                 // object head [256,20]
        else          out[20*NROI + roi*4 + (nh2 - 20)] = val; // motion head [256,4]
    }
}

extern "C" void kernel_launch(void* const* d_in, const int* in_sizes, int n_in,
                              void* d_out, int out_size, void* d_ws, size_t ws_size,
                              hipStream_t stream) {
    const float* feat  = (const float*)d_in[0];
    const float* bb    = (const float*)d_in[1];
    const float* w_obj = (const float*)d_in[2];
    const float* b_obj = (const float*)d_in[3];
    const float* w_mot = (const float*)d_in[4];
    const float* b_mot = (const float*)d_in[5];
    float* out    = (float*)d_out;
    float* pooled = (float*)d_ws;   // needs NROI*CK*4 = 13.1 MB

    prroi_pool_wmma<<<dim3(NROI), 256, 0, stream>>>(feat, bb, pooled);
    head_wmma<<<dim3(16, 2), 256, 0, stream>>>(pooled, w_obj, b_obj, w_mot, b_mot, out);
}